// interpolation_32710470926871
// MI455X (gfx1250) — compile-verified
//
#include <hip/hip_runtime.h>

typedef float v4f __attribute__((ext_vector_type(4)));
typedef float v2f __attribute__((ext_vector_type(2)));

namespace {
constexpr int kH  = 128;
constexpr int kW  = 128;
constexpr int kC  = 64;
constexpr int kHW = kH * kW;
constexpr int kP  = 8;                      // inter_points (reference: 8)
constexpr float kHalfLen = 7.0f / 128.0f;   // HALF_LENGTH / H
}

// One thread = one sampled point (all 64 channels).
// grid.y = batch, grid.x = point-block within batch -> no integer division.
// Wave32: 32 consecutive points share <=2 anchors -> corner gathers of a wave
// land in a ~15x15 pixel box per channel plane (great L1/L2 locality).
// Output is streamed with non-temporal 128-bit stores so the 33.5MB feature
// map stays resident in the 192MB L2 across the whole dispatch.
__global__ __launch_bounds__(256)
void bilerp_gather_kernel(const float* __restrict__ fm,
                          const float* __restrict__ anchor,
                          float* __restrict__ out,
                          int N, int M)
{
    const int b      = blockIdx.y;
    const int within = blockIdx.x * blockDim.x + threadIdx.x;
    if (within >= M) return;

    const int n   = within >> 6;            // anchor index (P*P == 64)
    const int rem = within & 63;
    // grid reshape order: m = ((n*P + i)*P + j); x uses t[i], y uses t[j]
    const float ti = (float)(rem >> 3) * (1.0f / 7.0f);
    const float tj = (float)(rem & 7)  * (1.0f / 7.0f);

    const v2f an = ((const v2f*)anchor)[(size_t)b * N + n];

    // x coordinate
    float lo = fminf(fmaxf(an.x - kHalfLen, 0.0f), 1.0f);
    float hi = fminf(fmaxf(an.x + kHalfLen, 0.0f), 1.0f);
    const float px = fminf(fmaxf((lo + (hi - lo) * ti) * 127.0f, 0.0f), 127.0f);
    // y coordinate
    lo = fminf(fmaxf(an.y - kHalfLen, 0.0f), 1.0f);
    hi = fminf(fmaxf(an.y + kHalfLen, 0.0f), 1.0f);
    const float py = fminf(fmaxf((lo + (hi - lo) * tj) * 127.0f, 0.0f), 127.0f);

    // Branch-free corners: x0=min(floor,126), dx=px-x0 reproduces floor/ceil
    // semantics exactly (dx==0 at integers <127; x0=126,dx=1 at px==127) and
    // guarantees x0+1 <= 127 (same for y) -> unguarded 4-corner loads.
    int x0 = (int)floorf(px); x0 = (x0 > 126) ? 126 : x0;
    int y0 = (int)floorf(py); y0 = (y0 > 126) ? 126 : y0;
    const float dx = px - (float)x0;
    const float dy = py - (float)y0;

    const float* __restrict__ src = fm + (size_t)b * kC * kHW + (size_t)(y0 * kW + x0);
    float* __restrict__ dst = out + ((size_t)b * M + within) * kC; // [B, M, C]

    #pragma unroll 4
    for (int c = 0; c < kC; c += 4) {
        const float* p0 = src + (size_t)c * kHW;
        if (c < kC - 4) {
            // next 4-channel chunk is +256KB away; scattered gathers defeat HW
            // prefetchers, so issue an explicit global_prefetch.
            __builtin_prefetch(p0 + 4 * (size_t)kHW, 0, 1);
        }
        v4f r;
        #pragma unroll
        for (int q = 0; q < 4; ++q) {
            const float* pc = p0 + (size_t)q * kHW;
            const float v00 = pc[0];            // (y0, x0)
            const float v01 = pc[1];            // (y0, x0+1)
            const float v10 = pc[kW];           // (y0+1, x0)
            const float v11 = pc[kW + 1];       // (y0+1, x0+1)
            const float top = __builtin_fmaf(dx, v01 - v00, v00);
            const float bot = __builtin_fmaf(dx, v11 - v10, v10);
            r[q] = __builtin_fmaf(dy, bot - top, top);
        }
        // 128-bit aligned streaming store (th:TH_STORE_NT): output is write-
        // once, keep it out of L2 so the feature map stays resident.
        __builtin_nontemporal_store(r, (v4f*)(dst + c));
    }
}

extern "C" void kernel_launch(void* const* d_in, const int* in_sizes, int n_in,
                              void* d_out, int out_size, void* d_ws, size_t ws_size,
                              hipStream_t stream) {
    const float* fm     = (const float*)d_in[0];
    const float* anchor = (const float*)d_in[1];
    float* out          = (float*)d_out;

    const int B = in_sizes[0] / (kC * kHW);            // 8
    const int N = (B > 0) ? in_sizes[1] / (2 * B) : 0; // 1024
    const int M = N * kP * kP;                         // 65536 points / batch
    if (B <= 0 || M <= 0) return;

    const int threads = 256;                           // 8 wave32 waves / block
    dim3 grid((M + threads - 1) / threads, B);
    bilerp_gather_kernel<<<grid, threads, 0, stream>>>(fm, anchor, out, N, M);
}